// KalmanFilter_45672682226124
// MI455X (gfx1250) — compile-verified
//
#include <hip/hip_runtime.h>
#include <hip/hip_bf16.h>

// Kalman filter on gfx1250: one wave32 per group, covariance carried in the
// 8-VGPR WMMA C-layout, all matrix products via V_WMMA_F32_16X16X4_F32.
// Layout conversions (C-layout <-> A/B operand layouts) go through per-wave
// LDS scratch; DS ops are in-order within a wave so no barriers in the loop.

typedef float v2f __attribute__((ext_vector_type(2)));
typedef float v4f __attribute__((ext_vector_type(4)));
typedef float v8f __attribute__((ext_vector_type(8)));

#define WAVES_PER_BLOCK 8

static __device__ __forceinline__ float rdl(float v, int l) {
  return __int_as_float(__builtin_amdgcn_readlane(__float_as_int(v), l));
}

// D = A(16x4) * B(4x16) + C(16x16), f32. 8-arg form per CDNA5 bridge doc.
static __device__ __forceinline__ v8f wmma4(v2f a, v2f b, v8f c) {
  return __builtin_amdgcn_wmma_f32_16x16x4_f32(false, a, false, b, (short)0, c,
                                               false, false);
}

__global__ __launch_bounds__(256, 1) void kalman_kernel(
    const float* __restrict__ inp,   // [G,T,4]
    const float* __restrict__ Fm,    // [16,16]
    const float* __restrict__ Qm,    // [16,16]
    const float* __restrict__ Hm,    // [4,16]
    const float* __restrict__ Rm,    // [4,4]
    const float* __restrict__ im,    // [G,16]
    const float* __restrict__ ic,    // [G,16,16]
    float* __restrict__ out, int G, int T) {
  // block-shared constants (row-major)
  __shared__ __align__(16) float sF[256];    // F
  __shared__ __align__(16) float sFT[256];   // F^T
  __shared__ __align__(16) float sHf[256];   // H padded to 16x16 (rows 4..15 = 0)
  __shared__ __align__(16) float sHfT[256];  // H^T padded (cols 4..15 = 0)
  // per-wave scratch
  __shared__ __align__(16) float sCov[WAVES_PER_BLOCK][256];  // P row-major
  __shared__ __align__(16) float sHP[WAVES_PER_BLOCK][256];   // HP, rows 4..15 zero
  __shared__ __align__(16) float sW[WAVES_PER_BLOCK][256];    // P_u
  __shared__ __align__(16) float sW2[WAVES_PER_BLOCK][256];   // T1 = F*P_u
  __shared__ __align__(16) float sK[WAVES_PER_BLOCK][64];     // -K, 16x4 row-major
  __shared__ __align__(16) float sSi[WAVES_PER_BLOCK][64];    // -S^-1, 4x16 padded

  const int tid = threadIdx.x;
  const int lane = tid & 31;
  const int w = tid >> 5;
  const int hi = lane >> 4;   // lane half
  const int n = lane & 15;    // column (B/C) / row (A)
  const int hi8 = hi << 3;
  const int g = blockIdx.x * WAVES_PER_BLOCK + w;

  {
    int r = tid >> 4, c = tid & 15;
    sF[tid] = Fm[tid];
    sFT[tid] = Fm[c * 16 + r];
    sHf[tid] = (r < 4) ? Hm[r * 16 + c] : 0.f;
    sHfT[tid] = (c < 4) ? Hm[c * 16 + r] : 0.f;
  }
  float* myCov = sCov[w];
  float* myHP = sHP[w];
  float* myW = sW[w];
  float* myW2 = sW2[w];
  float* myK = sK[w];
  float* mySi = sSi[w];
  for (int idx = lane; idx < 256; idx += 32)
    if (idx >= 64) myHP[idx] = 0.f;  // HP rows 4..15 stay zero forever
  for (int idx = lane; idx < 64; idx += 32) mySi[idx] = 0.f;  // cols 4..15 stay 0
  __syncthreads();
  if (g >= G) return;

  const long go = (long)g * T;
  float* out_means = out;                              // [G,T,16]
  float* out_covs = out + (long)G * T * 16;            // [G,T,16,16]
  float* out_mm = out_covs + (long)G * T * 256;        // [G,T,4]
  float* out_mc = out_mm + (long)G * T * 4;            // [G,T,4,4]

  // wave-resident WMMA C-layout matrices: row = r+hi8, col = n
  v8f covC, qC, rC;
#pragma unroll
  for (int r = 0; r < 8; ++r) {
    int row = r + hi8;
    covC[r] = ic[(long)g * 256 + row * 16 + n];
    qC[r] = Qm[row * 16 + n];
    rC[r] = (row < 4 && n < 4) ? Rm[row * 4 + n] : 0.f;
  }
  float vmean = (lane < 16) ? im[(long)g * 16 + lane] : 0.f;
  float Fcol[16], Hcol[16];
#pragma unroll
  for (int j = 0; j < 16; ++j) Fcol[j] = Fm[n * 16 + j];  // lane s: F[s][j]
#pragma unroll
  for (int s = 0; s < 16; ++s) Hcol[s] = (lane < 4) ? Hm[lane * 16 + s] : 0.f;

  // seed LDS P and outputs for t=0
#pragma unroll
  for (int r = 0; r < 8; ++r) {
    float v = covC[r];
    myCov[(r + hi8) * 16 + n] = v;
    out_covs[go * 256 + (r + hi8) * 16 + n] = v;
  }
  if (lane < 16) out_means[go * 16 + lane] = vmean;

  const v8f z8 = {0.f, 0.f, 0.f, 0.f, 0.f, 0.f, 0.f, 0.f};

#pragma unroll 1
  for (int i = 0; i < T; ++i) {
    // ---- meas mean: hm = H*mean (lanes 0..3) ----
    float hm = 0.f;
#pragma unroll
    for (int s = 0; s < 16; ++s) hm = fmaf(Hcol[s], rdl(vmean, s), hm);
    if (lane < 4) out_mm[(go + i) * 4 + lane] = hm;

    // ---- HP = Hf * P  (4 chained WMMAs over K) ----
    v8f hp = z8;
#pragma unroll
    for (int k = 0; k < 16; k += 4) {
      v2f a = *(const v2f*)&sHf[n * 16 + k + 2 * hi];
      v2f b;
      b.x = myCov[(k + 2 * hi) * 16 + n];
      b.y = myCov[(k + 1 + 2 * hi) * 16 + n];
      hp = wmma4(a, b, hp);
    }
    if (lane < 16) {
#pragma unroll
      for (int r = 0; r < 4; ++r) myHP[r * 16 + lane] = hp[r];
    }

    // ---- Smat = HP * H^T + R  (== meas cov for index i) ----
    v8f sm = rC;
#pragma unroll
    for (int k = 0; k < 16; k += 4) {
      v2f a = *(const v2f*)&myHP[n * 16 + k + 2 * hi];
      v2f b;
      b.x = sHfT[(k + 2 * hi) * 16 + n];
      b.y = sHfT[(k + 1 + 2 * hi) * 16 + n];
      sm = wmma4(a, b, sm);
    }
    if (lane < 4) {
#pragma unroll
      for (int r = 0; r < 4; ++r)
        out_mc[((go + i) * 4 + r) * 4 + lane] = sm[r];
    }

    if (i == T - 1) break;

    // ---- -S^-1 via broadcast + redundant 4x4 adjugate ----
    float a00 = rdl(sm[0], 0), a01 = rdl(sm[0], 1), a02 = rdl(sm[0], 2), a03 = rdl(sm[0], 3);
    float a10 = rdl(sm[1], 0), a11 = rdl(sm[1], 1), a12 = rdl(sm[1], 2), a13 = rdl(sm[1], 3);
    float a20 = rdl(sm[2], 0), a21 = rdl(sm[2], 1), a22 = rdl(sm[2], 2), a23 = rdl(sm[2], 3);
    float a30 = rdl(sm[3], 0), a31 = rdl(sm[3], 1), a32 = rdl(sm[3], 2), a33 = rdl(sm[3], 3);
    float s0 = a00 * a11 - a10 * a01, s1 = a00 * a12 - a10 * a02;
    float s2 = a00 * a13 - a10 * a03, s3 = a01 * a12 - a11 * a02;
    float s4 = a01 * a13 - a11 * a03, s5 = a02 * a13 - a12 * a03;
    float c5 = a22 * a33 - a32 * a23, c4 = a21 * a33 - a31 * a23;
    float c3 = a21 * a32 - a31 * a22, c2 = a20 * a33 - a30 * a23;
    float c1 = a20 * a32 - a30 * a22, c0 = a20 * a31 - a30 * a21;
    float det = s0 * c5 - s1 * c4 + s2 * c3 + s3 * c2 - s4 * c1 + s5 * c0;
    float nd = -1.0f / det;  // negative => WMMA produces -K directly
    v4f i0 = {( a11 * c5 - a12 * c4 + a13 * c3) * nd,
              (-a01 * c5 + a02 * c4 - a03 * c3) * nd,
              ( a31 * s5 - a32 * s4 + a33 * s3) * nd,
              (-a21 * s5 + a22 * s4 - a23 * s3) * nd};
    v4f i1 = {(-a10 * c5 + a12 * c2 - a13 * c1) * nd,
              ( a00 * c5 - a02 * c2 + a03 * c1) * nd,
              (-a30 * s5 + a32 * s2 - a33 * s1) * nd,
              ( a20 * s5 - a22 * s2 + a23 * s1) * nd};
    v4f i2 = {( a10 * c4 - a11 * c2 + a13 * c0) * nd,
              (-a00 * c4 + a01 * c2 - a03 * c0) * nd,
              ( a30 * s4 - a31 * s2 + a33 * s0) * nd,
              (-a20 * s4 + a21 * s2 - a23 * s0) * nd};
    v4f i3 = {(-a10 * c3 + a11 * c1 - a12 * c0) * nd,
              ( a00 * c3 - a01 * c1 + a02 * c0) * nd,
              (-a30 * s3 + a31 * s1 - a32 * s0) * nd,
              ( a20 * s3 - a21 * s1 + a22 * s0) * nd};
    if (lane == 0) {
      *(v4f*)&mySi[0] = i0;
      *(v4f*)&mySi[16] = i1;
      *(v4f*)&mySi[32] = i2;
      *(v4f*)&mySi[48] = i3;
    }

    // ---- nK = HP^T * (-S^-1)  (one 16x16x4 WMMA) ----
    v8f nk;
    {
      v2f a;  // A[m][k] = HP[k][m]
      a.x = myHP[(2 * hi) * 16 + n];
      a.y = myHP[(2 * hi + 1) * 16 + n];
      v2f b;  // rows of -S^-1 (padded cols read zero)
      b.x = mySi[(2 * hi) * 16 + n];
      b.y = mySi[(2 * hi + 1) * 16 + n];
      nk = wmma4(a, b, z8);
    }
    if (n < 4) {  // store -K row-major 16x4
#pragma unroll
      for (int r = 0; r < 8; ++r) myK[(r + hi8) * 4 + n] = nk[r];
    }

    // ---- mean update: mu = mean + K*resid = mean - nK*resid ----
    float vobs = (lane < 4) ? inp[(go + i) * 4 + lane] : 0.f;
    float resid = vobs - hm;
    float q0 = rdl(resid, 0), q1 = rdl(resid, 1), q2 = rdl(resid, 2), q3 = rdl(resid, 3);
    v4f krow = *(const v4f*)&myK[n * 4];  // lane s: -K[s][0..3]
    float mu = vmean - (krow.x * q0 + krow.y * q1 + krow.z * q2 + krow.w * q3);

    // ---- P_u = P + (-K)*HP  (one WMMA, C = P) ----
    v8f cu;
    {
      v2f a = *(const v2f*)&myK[n * 4 + 2 * hi];
      v2f b;
      b.x = myHP[(2 * hi) * 16 + n];
      b.y = myHP[(2 * hi + 1) * 16 + n];
      cu = wmma4(a, b, covC);
    }
#pragma unroll
    for (int r = 0; r < 8; ++r) myW[(r + hi8) * 16 + n] = cu[r];

    // ---- T1 = F * P_u ----
    v8f t1 = z8;
#pragma unroll
    for (int k = 0; k < 16; k += 4) {
      v2f a = *(const v2f*)&sF[n * 16 + k + 2 * hi];
      v2f b;
      b.x = myW[(k + 2 * hi) * 16 + n];
      b.y = myW[(k + 1 + 2 * hi) * 16 + n];
      t1 = wmma4(a, b, t1);
    }
#pragma unroll
    for (int r = 0; r < 8; ++r) myW2[(r + hi8) * 16 + n] = t1[r];

    // ---- P' = T1 * F^T + Q ----
    v8f cp = qC;
#pragma unroll
    for (int k = 0; k < 16; k += 4) {
      v2f a = *(const v2f*)&myW2[n * 16 + k + 2 * hi];
      v2f b;
      b.x = sFT[(k + 2 * hi) * 16 + n];
      b.y = sFT[(k + 1 + 2 * hi) * 16 + n];
      cp = wmma4(a, b, cp);
    }
    covC = cp;

    // ---- mean' = F * mu ----
    float mp = 0.f;
#pragma unroll
    for (int j = 0; j < 16; ++j) mp = fmaf(Fcol[j], rdl(mu, j), mp);
    vmean = mp;

    // ---- emit prediction i+1, refresh LDS P ----
    if (lane < 16) out_means[(go + i + 1) * 16 + lane] = vmean;
#pragma unroll
    for (int r = 0; r < 8; ++r) {
      float v = covC[r];
      out_covs[(go + i + 1) * 256 + (r + hi8) * 16 + n] = v;
      myCov[(r + hi8) * 16 + n] = v;
    }
  }
}

extern "C" void kernel_launch(void* const* d_in, const int* in_sizes, int n_in,
                              void* d_out, int out_size, void* d_ws, size_t ws_size,
                              hipStream_t stream) {
  const float* inp = (const float*)d_in[0];
  const float* Fm = (const float*)d_in[1];
  const float* Qm = (const float*)d_in[2];
  const float* Hm = (const float*)d_in[3];
  const float* Rm = (const float*)d_in[4];
  const float* im = (const float*)d_in[5];
  const float* ic = (const float*)d_in[6];
  int G = in_sizes[5] / 16;          // init_mean is [G,16]
  int T = in_sizes[0] / (G * 4);     // input is [G,T,4]
  dim3 grid((G + WAVES_PER_BLOCK - 1) / WAVES_PER_BLOCK), block(256);
  kalman_kernel<<<grid, block, 0, stream>>>(inp, Fm, Qm, Hm, Rm, im, ic,
                                            (float*)d_out, G, T);
}